// ExponentialDecay_10385230922331
// MI455X (gfx1250) — compile-verified
//
#include <hip/hip_runtime.h>

typedef float v2f __attribute__((ext_vector_type(2)));
typedef float v8f __attribute__((ext_vector_type(8)));

#define TN 4            // column tiles per wave
#define WAVES_PER_BLOCK 8

// env[b,n] = r_b^n = exp2( log2(r_b) * n ), computed as a rank-1 WMMA outer
// product (v_wmma_f32_16x16x4_f32) followed by per-element v_exp_f32, NT-stored.
template <bool GUARD>
__global__ __launch_bounds__(256) void ExponentialDecay_wmma_kernel(
    const float* __restrict__ decay, float* __restrict__ out,
    int B, int N, int groupsN, int totalGroups) {
  const int lane = threadIdx.x & 31;
  const int wave = blockIdx.x * WAVES_PER_BLOCK + (threadIdx.x >> 5);
  if (wave >= totalGroups) return;           // wave-uniform: EXEC stays all-1s

  const int tm = wave / groupsN;             // 16-row tile index
  const int gn = wave % groupsN;             // group of TN 16-col tiles

  const int m  = lane & 15;
  const bool lo = lane < 16;
  const int row = tm * 16 + m;

  // log2(r) for this lane's row (lanes 0-15 carry K=0 of the A matrix)
  float logr = 0.0f;
  if (lo) {
    int ri = (!GUARD || row < B) ? row : (B - 1);
    float dv = decay[ri];
    float decay_ms      = 10.0f + (2000.0f - 10.0f) * dv;
    float decay_samples = decay_ms * 48.0f;          // * SAMPLE_RATE / 1000
    float x = 6.91f / decay_samples;                 // r = 1 - x
    logr = log1pf(-x) * 1.44269504088896340736f;     // log2(r)
  }

  // A (16x4): column K=0 = log2(r), rest zero.
  v2f a;
  a.x = lo ? logr : 0.0f;   // lanes 0-15: K=0 ; lanes 16-31: K=2 (zero)
  a.y = 0.0f;               // K=1 / K=3

  const int colBase = gn * (TN * 16);
  float ncol = (float)(colBase + m);

  // Incremental store pointer: this lane covers rows [tm*16 (+8 hi half) .. +7],
  // column colBase+m; consecutive D VGPRs are consecutive rows (stride N floats).
  float* p0 = out + (size_t)(tm * 16 + (lo ? 0 : 8)) * (size_t)N
                  + (size_t)(colBase + m);

  #pragma unroll
  for (int t = 0; t < TN; ++t) {
    // B (4x16): row K=0 = column index, rest zero.
    v2f bm;
    bm.x = lo ? ncol : 0.0f;  // lanes 0-15: K=0 ; lanes 16-31: K=2 (zero)
    bm.y = 0.0f;              // K=1 / K=3
    v8f c = {};
    // D[mrow][ncol] = log2(r_mrow) * ncol
    v8f acc = __builtin_amdgcn_wmma_f32_16x16x4_f32(
        /*neg_a=*/false, a, /*neg_b=*/false, bm,
        /*c_mod=*/(short)0, c, /*reuse_a=*/false, /*reuse_b=*/false);

    float* p = p0;
    const int col = colBase + t * 16 + m;
    if (!GUARD || col < N) {
      #pragma unroll
      for (int v = 0; v < 8; ++v) {
        if (!GUARD || (tm * 16 + v + (lo ? 0 : 8)) < B) {
          float val = __builtin_amdgcn_exp2f(acc[v]); // v_exp_f32: 2^(n*log2 r)
          // 196.6 MB output > 192 MB L2, never re-read -> non-temporal store
          __builtin_nontemporal_store(val, p);
        }
        p += N;                                       // next row, one 64-bit add
      }
    }
    p0 += 16;                                         // next 16-col tile
    ncol += 16.0f;
  }
}

extern "C" void kernel_launch(void* const* d_in, const int* in_sizes, int n_in,
                              void* d_out, int out_size, void* d_ws, size_t ws_size,
                              hipStream_t stream) {
  // inputs: d_in[0] = num_samples (int scalar, on device), d_in[1] = decay [B,1] f32
  const float* decay = (const float*)d_in[1];
  const int B = in_sizes[1];                 // 512
  const int N = out_size / B;                // num_samples (96000), derived on host
  float* out = (float*)d_out;

  const int tilesM = (B + 15) / 16;
  const int tilesN = (N + 15) / 16;
  const int groupsN = (tilesN + TN - 1) / TN;
  const int totalGroups = tilesM * groupsN;
  const int blocks = (totalGroups + WAVES_PER_BLOCK - 1) / WAVES_PER_BLOCK;

  const bool aligned = (B % 16 == 0) && (N % (TN * 16) == 0);
  if (aligned) {
    ExponentialDecay_wmma_kernel<false><<<blocks, WAVES_PER_BLOCK * 32, 0, stream>>>(
        decay, out, B, N, groupsN, totalGroups);
  } else {
    ExponentialDecay_wmma_kernel<true><<<blocks, WAVES_PER_BLOCK * 32, 0, stream>>>(
        decay, out, B, N, groupsN, totalGroups);
  }
}